// SE3ClassificationModel_4939212391242
// MI455X (gfx1250) — compile-verified
//
#include <hip/hip_runtime.h>
#include <math.h>

#define B_    4
#define N_    640
#define DIM_  32
#define HEADS_ 4
#define DH_   8
#define RHID_ 16
#define NCLS_ 10
#define EPS_  1e-8f
#define SQRT3_ 1.7320508075688772f

typedef __attribute__((ext_vector_type(2))) float v2f;
typedef __attribute__((ext_vector_type(8))) float v8f;

// ---------------------------------------------------------------------------
// feats[b,n,c] = features[b,n]*proj_w[c] + proj_b[c]
__global__ void k_proj(const float* __restrict__ features, const float* __restrict__ pw,
                       const float* __restrict__ pb, float* __restrict__ feats) {
    int idx = blockIdx.x * 256 + threadIdx.x;       // B*N*DIM = 81920
    int c = idx & 31;
    int row = idx >> 5;
    feats[idx] = features[row] * pw[c] + pb[c];
}

// ---------------------------------------------------------------------------
// LayerNorm over DIM=32; one row per 32-lane wave slice (wave32)
__global__ void k_ln(const float* __restrict__ feats, const float* __restrict__ g,
                     const float* __restrict__ b, float* __restrict__ hn) {
    int lane = threadIdx.x & 31;
    int row = blockIdx.x * 8 + (threadIdx.x >> 5);
    float x = feats[row * 32 + lane];
    float s = x;
    for (int m = 16; m; m >>= 1) s += __shfl_xor(s, m, 32);
    float mean = s * (1.f / 32.f);
    float d = x - mean;
    float s2 = d * d;
    for (int m = 16; m; m >>= 1) s2 += __shfl_xor(s2, m, 32);
    float var = s2 * (1.f / 32.f);
    hn[row * 32 + lane] = d * rsqrtf(var + 1e-5f) * g[lane] + b[lane];
}

// ---------------------------------------------------------------------------
// out[2560 x 32] = A[2560 x K] @ W[K x 32] (+bias)(+resid), f32 WMMA 16x16x4.
// One wave per 16x16 output tile. tiles = 160 mtiles * 2 ntiles = 320 -> grid 40 x 256.
__global__ void k_gemm_wmma(const float* __restrict__ A, const float* __restrict__ W,
                            const float* __restrict__ bias, const float* __restrict__ resid,
                            float* __restrict__ out, int K) {
    int wave = threadIdx.x >> 5;
    int lane = threadIdx.x & 31;
    int tile = blockIdx.x * 8 + wave;
    int mtile = tile >> 1;
    int ntile = tile & 1;
    int half = lane >> 4;          // which half-wave (K split)
    int l16  = lane & 15;
    int arow = mtile * 16 + l16;   // A: M = lane&15 for both halves
    int bcol = ntile * 16 + l16;   // B: N = lane&15 for both halves
    v8f c = {};
    for (int k0 = 0; k0 < K; k0 += 4) {
        int kk = k0 + half * 2;    // half 0 -> K={k0,k0+1}, half 1 -> K={k0+2,k0+3}
        v2f a, bf;
        a.x  = A[arow * K + kk];
        a.y  = A[arow * K + kk + 1];
        bf.x = W[kk * DIM_ + bcol];
        bf.y = W[(kk + 1) * DIM_ + bcol];
        c = __builtin_amdgcn_wmma_f32_16x16x4_f32(false, a, false, bf,
                                                  (short)0, c, false, false);
    }
    int rbase = mtile * 16 + half * 8;   // D: VGPR r -> M=r (lanes 0-15), M=r+8 (16-31)
    #pragma unroll
    for (int r = 0; r < 8; ++r) {
        float val = c[r];
        if (bias)  val += bias[bcol];
        int o = (rbase + r) * DIM_ + bcol;
        if (resid) val += resid[o];
        out[o] = val;
    }
}

// ---------------------------------------------------------------------------
// Fully fused: logits -> masked softmax -> (basis + radial MLP) -> deg-0/1/2
// messages -> invariant norms -> cat[b,n,96].  One block per (b,i).
__global__ void k_attn_msg(const float* __restrict__ coords,
                           const float* __restrict__ q, const float* __restrict__ k,
                           const float* __restrict__ v,
                           const float* __restrict__ rw1, const float* __restrict__ rb1,
                           const float* __restrict__ rw2, const float* __restrict__ rb2,
                           float* __restrict__ cat) {
    __shared__ float s_qi[32];
    __shared__ float s_att[HEADS_ * N_];   // 2560
    __shared__ float s_red[256];
    __shared__ float s_coef[128 * 36];     // per-edge 4 heads x 9 coefficients
    __shared__ float s_vt[128 * 32];       // V tile
    __shared__ float s_acc[8 * 32 * 9];    // jp x (h,d) x m partials

    int t  = threadIdx.x;
    int bi = blockIdx.x;          // b*N + i
    int b  = bi / N_;
    int i  = bi - b * N_;

    if (t < 32) s_qi[t] = q[bi * 32 + t];
    __syncthreads();

    int h  = t >> 6;              // 0..3
    int jl = t & 63;
    const float scale = 0.35355339059327373f;   // 1/sqrt(8)
    const float* kb = k + (size_t)b * N_ * 32 + h * 8;

    // ---- logits (diag masked to -1e9, matching reference order) ----
    for (int j = jl; j < N_; j += 64) {
        const float* kr = kb + j * 32;
        float s = 0.f;
        #pragma unroll
        for (int d = 0; d < 8; ++d) s += s_qi[h * 8 + d] * kr[d];
        s *= scale;
        if (j == i) s = -1e9f;
        s_att[h * N_ + j] = s;
    }
    __syncthreads();

    // ---- softmax over j (64 threads per head) ----
    float lmax = -3.0e38f;
    for (int j = jl; j < N_; j += 64) lmax = fmaxf(lmax, s_att[h * N_ + j]);
    s_red[t] = lmax;
    __syncthreads();
    for (int sft = 32; sft > 0; sft >>= 1) {
        if (jl < sft) s_red[t] = fmaxf(s_red[t], s_red[t + sft]);
        __syncthreads();
    }
    float mx = s_red[h * 64];
    __syncthreads();
    float lsum = 0.f;
    for (int j = jl; j < N_; j += 64) {
        float e = __expf(s_att[h * N_ + j] - mx);
        s_att[h * N_ + j] = e;
        lsum += e;
    }
    s_red[t] = lsum;
    __syncthreads();
    for (int sft = 32; sft > 0; sft >>= 1) {
        if (jl < sft) s_red[t] += s_red[t + sft];
        __syncthreads();
    }
    float inv = 1.f / s_red[h * 64];
    __syncthreads();
    for (int j = jl; j < N_; j += 64) s_att[h * N_ + j] *= inv;
    __syncthreads();

    // ---- messages over 5 tiles of 128 edges ----
    float cx = coords[bi * 3 + 0];
    float cy = coords[bi * 3 + 1];
    float cz = coords[bi * 3 + 2];
    float acc[9];
    #pragma unroll
    for (int m = 0; m < 9; ++m) acc[m] = 0.f;
    int hd = t & 31, jp = t >> 5;
    int hh = hd >> 3, dd = hd & 7;

    for (int jt = 0; jt < 5; ++jt) {
        int j0 = jt * 128;
        if (t < 128) {
            int j = j0 + t;
            float rx = cx - coords[(b * N_ + j) * 3 + 0];
            float ry = cy - coords[(b * N_ + j) * 3 + 1];
            float rz = cz - coords[(b * N_ + j) * 3 + 2];
            float dist = sqrtf(rx * rx + ry * ry + rz * rz + EPS_);
            float idn = 1.f / dist;
            float ux = rx * idn, uy = ry * idn, uz = rz * idn;
            float y2_0 = SQRT3_ * ux * uy;
            float y2_1 = SQRT3_ * uy * uz;
            float y2_2 = 0.5f * (3.f * uz * uz - 1.f);
            float y2_3 = SQRT3_ * ux * uz;
            float y2_4 = 0.5f * SQRT3_ * (ux * ux - uy * uy);
            float g12[12];
            #pragma unroll
            for (int o = 0; o < 12; ++o) g12[o] = rb2[o];
            #pragma unroll
            for (int r = 0; r < RHID_; ++r) {
                float rh = fmaxf(dist * rw1[r] + rb1[r], 0.f);
                #pragma unroll
                for (int o = 0; o < 12; ++o) g12[o] += rh * rw2[r * 12 + o];
            }
            #pragma unroll
            for (int hx = 0; hx < 4; ++hx) {
                float a  = s_att[hx * N_ + j];
                float c0 = a * g12[hx * 3 + 0];
                float c1 = a * g12[hx * 3 + 1];
                float c2 = a * g12[hx * 3 + 2];
                float* cf = &s_coef[t * 36 + hx * 9];
                cf[0] = c0;
                cf[1] = c1 * ux; cf[2] = c1 * uy; cf[3] = c1 * uz;
                cf[4] = c2 * y2_0; cf[5] = c2 * y2_1; cf[6] = c2 * y2_2;
                cf[7] = c2 * y2_3; cf[8] = c2 * y2_4;
            }
        } else {
            int e = t - 128;
            const float* vr = v + (size_t)(b * N_ + j0 + e) * 32;
            #pragma unroll
            for (int c = 0; c < 32; ++c) s_vt[e * 32 + c] = vr[c];
        }
        __syncthreads();
        for (int e = jp; e < 128; e += 8) {
            float vv = s_vt[e * 32 + hh * 8 + dd];
            const float* cf = &s_coef[e * 36 + hh * 9];
            #pragma unroll
            for (int m = 0; m < 9; ++m) acc[m] += cf[m] * vv;
        }
        __syncthreads();
    }

    // ---- reduce partials across jp, build cat = [out0 | inv1 | inv2] per head ----
    #pragma unroll
    for (int m = 0; m < 9; ++m) s_acc[(jp * 32 + hd) * 9 + m] = acc[m];
    __syncthreads();
    if (t < 32) {
        float s[9];
        #pragma unroll
        for (int m = 0; m < 9; ++m) {
            float x = 0.f;
            #pragma unroll
            for (int p = 0; p < 8; ++p) x += s_acc[(p * 32 + t) * 9 + m];
            s[m] = x;
        }
        int hx = t >> 3, dx = t & 7;
        float* cr = cat + (size_t)bi * 96 + hx * 24;
        cr[dx]      = s[0];
        cr[8 + dx]  = sqrtf(s[1] * s[1] + s[2] * s[2] + s[3] * s[3] + EPS_);
        cr[16 + dx] = sqrtf(s[4] * s[4] + s[5] * s[5] + s[6] * s[6] +
                            s[7] * s[7] + s[8] * s[8] + EPS_);
    }
}

// ---------------------------------------------------------------------------
// max-pool over N, then 32->64 relu MLP, then 64->10. Single block.
__global__ void k_head(const float* __restrict__ feats, const float* __restrict__ w1,
                       const float* __restrict__ b1, const float* __restrict__ w2,
                       const float* __restrict__ b2, float* __restrict__ out) {
    __shared__ float pool[4 * 32];
    __shared__ float hm[4 * 64];
    int t = threadIdx.x;
    if (t < 128) {
        int b = t >> 5, c = t & 31;
        float m = -3.0e38f;
        const float* f = feats + (size_t)b * N_ * 32 + c;
        for (int n = 0; n < N_; ++n) m = fmaxf(m, f[n * 32]);
        pool[t] = m;
    }
    __syncthreads();
    {
        int b = t >> 6, o = t & 63;
        float s = b1[o];
        #pragma unroll
        for (int c = 0; c < 32; ++c) s += pool[b * 32 + c] * w1[c * 64 + o];
        hm[t] = fmaxf(s, 0.f);
    }
    __syncthreads();
    if (t < 40) {
        int b = t / 10, cls = t - b * 10;
        float s = b2[cls];
        #pragma unroll
        for (int o = 0; o < 64; ++o) s += hm[b * 64 + o] * w2[o * 10 + cls];
        out[b * 10 + cls] = s;
    }
}

// ---------------------------------------------------------------------------
extern "C" void kernel_launch(void* const* d_in, const int* in_sizes, int n_in,
                              void* d_out, int out_size, void* d_ws, size_t ws_size,
                              hipStream_t stream) {
    (void)in_sizes; (void)n_in; (void)out_size; (void)ws_size;
    const float* features = (const float*)d_in[0];
    const float* coords   = (const float*)d_in[1];
    const float* proj_w   = (const float*)d_in[2];
    const float* proj_b   = (const float*)d_in[3];
    const float* ln_g     = (const float*)d_in[4];
    const float* ln_b     = (const float*)d_in[5];
    const float* wq       = (const float*)d_in[6];
    const float* wk       = (const float*)d_in[7];
    const float* wv       = (const float*)d_in[8];
    const float* rw1      = (const float*)d_in[9];
    const float* rb1      = (const float*)d_in[10];
    const float* rw2      = (const float*)d_in[11];
    const float* rb2      = (const float*)d_in[12];
    const float* wo       = (const float*)d_in[13];
    const float* bo       = (const float*)d_in[14];
    const float* fc1_w    = (const float*)d_in[15];
    const float* fc1_b    = (const float*)d_in[16];
    const float* fc2_w    = (const float*)d_in[17];
    const float* fc2_b    = (const float*)d_in[18];
    float* out = (float*)d_out;

    float* ws = (float*)d_ws;
    const size_t RC = (size_t)B_ * N_ * DIM_;      // 81920
    float* feats = ws;
    float* hnorm = feats + RC;
    float* q     = hnorm + RC;
    float* k     = q + RC;
    float* v     = k + RC;
    float* cat   = v + RC;                          // B*N*96

    k_proj<<<320, 256, 0, stream>>>(features, proj_w, proj_b, feats);
    for (int l = 0; l < 4; ++l) {
        k_ln<<<320, 256, 0, stream>>>(feats, ln_g + l * 32, ln_b + l * 32, hnorm);
        k_gemm_wmma<<<40, 256, 0, stream>>>(hnorm, wq + l * 1024, nullptr, nullptr, q, 32);
        k_gemm_wmma<<<40, 256, 0, stream>>>(hnorm, wk + l * 1024, nullptr, nullptr, k, 32);
        k_gemm_wmma<<<40, 256, 0, stream>>>(hnorm, wv + l * 1024, nullptr, nullptr, v, 32);
        k_attn_msg<<<2560, 256, 0, stream>>>(coords, q, k, v,
                                             rw1 + l * 16, rb1 + l * 16,
                                             rw2 + l * 192, rb2 + l * 12, cat);
        // feats = feats + cat @ wo + bo   (in-place residual, per-element RAW only)
        k_gemm_wmma<<<40, 256, 0, stream>>>(cat, wo + l * 3072, bo + l * 32, feats, feats, 96);
    }
    k_head<<<1, 256, 0, stream>>>(feats, fc1_w, fc1_b, fc2_w, fc2_b, out);
}